// ChamferLoss_34505767256624
// MI455X (gfx1250) — compile-verified
//
#include <hip/hip_runtime.h>

typedef __attribute__((ext_vector_type(2))) float v2f;
typedef __attribute__((ext_vector_type(8))) float v8f;

#define P_PTS 4096
#define NB 8
#define TILES (P_PTS / 16)   // 256 y-tiles per batch
#define CHUNK 128            // y-tiles staged in LDS per pass (32 KB)
#define BIGF 3.0e38f

__global__ void chamfer_zero_out(float* out) {
    if (threadIdx.x == 0) out[0] = 0.0f;
}

// One direction of chamfer: for each src point, min squared distance to valid tgt points.
// Block: 256 threads = 8 waves; blockIdx.y = batch, each wave owns one 16-row src tile.
__global__ __launch_bounds__(256)
void chamfer_dir_kernel(const float* __restrict__ src, const float* __restrict__ tgt,
                        const int* __restrict__ src_len, const int* __restrict__ tgt_len,
                        float* __restrict__ out)
{
    __shared__ v2f sB[CHUNK * 32];   // B-matrix operands for CHUNK tiles, one v2f per lane

    const int n        = blockIdx.y;
    const int lane     = threadIdx.x & 31;
    const int wave     = threadIdx.x >> 5;
    const int tileM    = blockIdx.x * 8 + wave;
    const int row_base = tileM * 16;
    const int tl       = tgt_len[n];

    const float* srcb = src + (size_t)n * P_PTS * 3;
    const float* tgtb = tgt + (size_t)n * P_PTS * 3;

    // ---- A matrix (16x4 f32): row m = [x0, x1, x2, 1.0]
    // lanes 0-15 hold K=0,1 ; lanes 16-31 hold K=2,3 (per ISA 32-bit A layout)
    v2f a;
    {
        const int p = row_base + (lane & 15);
        const float* sp = srcb + p * 3;
        if (lane < 16) { a[0] = sp[0]; a[1] = sp[1]; }
        else           { a[0] = sp[2]; a[1] = 1.0f; }
    }

    // running min of (-2 x.y + |y|^2) per D-matrix register
    v8f mn;
#pragma unroll
    for (int i = 0; i < 8; ++i) mn[i] = BIGF;

    for (int h = 0; h < TILES / CHUNK; ++h) {
        // ---- cooperatively stage B operands for CHUNK y-tiles into LDS
        // B (4x16 f32): lanes 0-15 hold K=0,1 ; lanes 16-31 hold K=2,3
        for (int e = threadIdx.x; e < CHUNK * 32; e += 256) {
            const int l = e & 31;
            const int t = (e >> 5) + h * CHUNK;
            const int p = t * 16 + (l & 15);
            const float* tp = tgtb + p * 3;
            v2f b;
            if (l < 16) {
                b[0] = -2.0f * tp[0];
                b[1] = -2.0f * tp[1];
            } else {
                const float y0 = tp[0], y1 = tp[1], y2 = tp[2];
                b[0] = -2.0f * y2;
                b[1] = (p < tl) ? (y0 * y0 + y1 * y1 + y2 * y2) : BIGF; // poison padded cols
            }
            sB[e] = b;
        }
        __syncthreads();

        // ---- hot loop: ds_load_b64 + v_wmma_f32_16x16x4_f32 + 8x v_min
#pragma unroll 4
        for (int t = 0; t < CHUNK; ++t) {
            v2f b = sB[t * 32 + lane];
            v8f c = {};
            v8f d = __builtin_amdgcn_wmma_f32_16x16x4_f32(
                false, a, false, b, (short)0, c, false, false);
#pragma unroll
            for (int i = 0; i < 8; ++i) mn[i] = fminf(mn[i], d[i]);
        }
        __syncthreads();
    }

    // ---- min across the 16 columns: shuffle tree within each 16-lane half
#pragma unroll
    for (int i = 0; i < 8; ++i) {
        float v = mn[i];
        v = fminf(v, __shfl_xor(v, 1, 32));
        v = fminf(v, __shfl_xor(v, 2, 32));
        v = fminf(v, __shfl_xor(v, 4, 32));
        v = fminf(v, __shfl_xor(v, 8, 32));
        mn[i] = v;
    }

    // lane 0 holds rows row_base+0..7, lane 16 holds rows row_base+8..15
    if (lane == 0 || lane == 16) {
        const int sl   = src_len[n];
        const int base = row_base + ((lane >= 16) ? 8 : 0);
        float s = 0.0f;
#pragma unroll
        for (int i = 0; i < 8; ++i) {
            const int row = base + i;
            if (row < sl) {
                const float* sp = srcb + row * 3;
                const float x2 = sp[0] * sp[0] + sp[1] * sp[1] + sp[2] * sp[2];
                s += x2 + mn[i];  // |x|^2 + min(-2x.y + |y|^2)
            }
        }
        atomicAdd(out, s / ((float)sl * (float)NB));
    }
}

extern "C" void kernel_launch(void* const* d_in, const int* in_sizes, int n_in,
                              void* d_out, int out_size, void* d_ws, size_t ws_size,
                              hipStream_t stream) {
    const float* x  = (const float*)d_in[0];
    const float* y  = (const float*)d_in[1];
    const int*   xl = (const int*)d_in[2];
    const int*   yl = (const int*)d_in[3];
    float* out = (float*)d_out;

    chamfer_zero_out<<<dim3(1), dim3(64), 0, stream>>>(out);

    dim3 grid(TILES / 8, NB);  // 32 x-tile groups x 8 batches
    dim3 block(256);
    // x -> y direction
    chamfer_dir_kernel<<<grid, block, 0, stream>>>(x, y, xl, yl, out);
    // y -> x direction
    chamfer_dir_kernel<<<grid, block, 0, stream>>>(y, x, yl, xl, out);
}